// GraphAttnGATAggregation_66331474920030
// MI455X (gfx1250) — compile-verified
//
#include <hip/hip_runtime.h>
#include <hip/hip_bf16.h>

#define N_NODES 50000
#define N_EDGES 800000
#define DIM 256
#define NHEAD 8
#define NCOL 16          // 8 cols s_u + 8 cols s_v
#define NTILES (N_NODES / 16)   // 3125, exact

typedef float v2f __attribute__((ext_vector_type(2)));
typedef float v8f __attribute__((ext_vector_type(8)));

// ---------------------------------------------------------------------------
// Monotonic-bits float atomic max (works for mixed signs; init with -inf).
// ---------------------------------------------------------------------------
__device__ __forceinline__ void atomicMaxFloat(float* addr, float val) {
    if (val >= 0.0f) {
        atomicMax((int*)addr, __float_as_int(val));
    } else {
        atomicMin((unsigned int*)addr, __float_as_uint(val));
    }
}

// ---------------------------------------------------------------------------
// Kernel 0: initialize out = 0, m = -inf, denom = 0
// ---------------------------------------------------------------------------
__global__ void gat_init_kernel(float* __restrict__ out,
                                float* __restrict__ m,
                                float* __restrict__ denom) {
    const int stride = gridDim.x * blockDim.x;
    const int base   = blockIdx.x * blockDim.x + threadIdx.x;
    for (int i = base; i < N_NODES * DIM; i += stride) out[i] = 0.0f;
    for (int i = base; i < N_NODES * NHEAD; i += stride) {
        m[i]     = -__builtin_inff();
        denom[i] = 0.0f;
    }
}

// ---------------------------------------------------------------------------
// Kernel 1: s[N,16] = x @ [W_u; W_v]^T (+ b_u on first 8 cols)
// One wave32 per 16-row tile; V_WMMA_F32_16X16X4_F32, K=256 in steps of 4.
// B (256x16) staged in LDS once per block.
// ---------------------------------------------------------------------------
__global__ __launch_bounds__(256) void gat_score_gemm_kernel(
        const float* __restrict__ x,
        const float* __restrict__ Wu,   // [8,256]
        const float* __restrict__ bu,   // [8]
        const float* __restrict__ Wv,   // [8,256]
        float* __restrict__ s) {        // [N,16]
    __shared__ float Bs[DIM * NCOL];    // Bs[k*16 + n] = B[k][n], 16 KB

    const int t = threadIdx.x;          // 0..255 -> k index for staging
    #pragma unroll
    for (int n = 0; n < NCOL; ++n) {
        float w = (n < NHEAD) ? Wu[n * DIM + t] : Wv[(n - NHEAD) * DIM + t];
        Bs[t * NCOL + n] = w;
    }
    __syncthreads();

    const int wave = t >> 5;            // 8 waves per block
    const int lane = t & 31;
    const int half = lane >> 4;         // 0: K pair (0,1); 1: K pair (2,3)
    const int mn   = lane & 15;         // row (A/C) or col (B/C) index

    const int tile = blockIdx.x * 8 + wave;
    if (tile >= NTILES) return;         // wave-uniform: EXEC stays all-ones

    const float* xrow = x + (size_t)tile * 16 * DIM;

    v8f c = {};
    for (int k0 = 0; k0 < DIM; k0 += 4) {
        const int k = k0 + 2 * half;
        v2f a, b;
        a.x = xrow[(size_t)mn * DIM + k];       // A[M=mn][K=k]
        a.y = xrow[(size_t)mn * DIM + k + 1];   // A[M=mn][K=k+1]
        b.x = Bs[k * NCOL + mn];                // B[K=k][N=mn]
        b.y = Bs[(k + 1) * NCOL + mn];          // B[K=k+1][N=mn]
        c = __builtin_amdgcn_wmma_f32_16x16x4_f32(
                /*neg_a=*/false, a, /*neg_b=*/false, b,
                /*c_mod=*/(short)0, c, /*reuse_a=*/false, /*reuse_b=*/false);
    }

    float bias = bu[mn & 7];
    if (mn >= NHEAD) bias = 0.0f;       // bias only on s_u columns

    #pragma unroll
    for (int r = 0; r < 8; ++r) {       // C VGPR r -> row r (+8 for lanes>=16)
        const int row = tile * 16 + r + 8 * half;
        s[(size_t)row * NCOL + mn] = c[r] + bias;
    }
}

// ---------------------------------------------------------------------------
// Kernel 2: e = LeakyReLU(s_u[src] + s_v[dst]); segment max into m[dst]
// One thread per (edge, head).
// ---------------------------------------------------------------------------
__global__ void gat_edge_logits_kernel(const float* __restrict__ s,
                                       const int* __restrict__ src,
                                       const int* __restrict__ dst,
                                       float* __restrict__ e_ws,
                                       float* __restrict__ m) {
    const int stride = gridDim.x * blockDim.x;
    for (int i = blockIdx.x * blockDim.x + threadIdx.x;
         i < N_EDGES * NHEAD; i += stride) {
        const int edge = i >> 3;
        const int h    = i & 7;
        const int sN = src[edge];
        const int dN = dst[edge];
        float e = s[(size_t)sN * NCOL + h] + s[(size_t)dN * NCOL + NHEAD + h];
        e = (e >= 0.0f) ? e : 0.2f * e;          // LeakyReLU(0.2)
        e_ws[i] = e;
        atomicMaxFloat(&m[dN * NHEAD + h], e);
    }
}

// ---------------------------------------------------------------------------
// Kernel 3: e_exp = exp(e - m[dst]); segment sum into denom[dst] (in-place)
// ---------------------------------------------------------------------------
__global__ void gat_edge_exp_kernel(const int* __restrict__ dst,
                                    float* __restrict__ e_ws,
                                    const float* __restrict__ m,
                                    float* __restrict__ denom) {
    const int stride = gridDim.x * blockDim.x;
    for (int i = blockIdx.x * blockDim.x + threadIdx.x;
         i < N_EDGES * NHEAD; i += stride) {
        const int edge = i >> 3;
        const int h    = i & 7;
        const int dN   = dst[edge];
        const float ex = __expf(e_ws[i] - m[dN * NHEAD + h]);
        e_ws[i] = ex;
        atomicAdd(&denom[dN * NHEAD + h], ex);
    }
}

// ---------------------------------------------------------------------------
// Kernel 4: out[dst] += x[src] * probs  (feature d belongs to head d & 7,
// per the reference's reshape(N, head_dim, num_heads)).
// One wave32 per edge; lane l handles features l, l+32, ... so head = l & 7
// is loop-invariant -> softmax prob computed once per lane.
// ---------------------------------------------------------------------------
__global__ __launch_bounds__(256) void gat_scatter_kernel(
        const float* __restrict__ x,
        const int* __restrict__ src,
        const int* __restrict__ dst,
        const float* __restrict__ e_exp,
        const float* __restrict__ denom,
        float* __restrict__ out) {
    const int lane   = threadIdx.x & 31;
    const int h      = lane & 7;
    const int wave   = blockIdx.x * (blockDim.x >> 5) + (threadIdx.x >> 5);
    const int nwaves = gridDim.x * (blockDim.x >> 5);

    for (int e = wave; e < N_EDGES; e += nwaves) {
        const int sN = src[e];
        const int dN = dst[e];

        // prefetch next edge's source row (emits global_prefetch_b8)
        const int e2 = e + nwaves;
        if (e2 < N_EDGES) {
            __builtin_prefetch(x + (size_t)src[e2] * DIM + lane * 8, 0, 0);
        }

        const float p = e_exp[(size_t)e * NHEAD + h] / denom[dN * NHEAD + h];
        const float* xr  = x   + (size_t)sN * DIM;
        float*       orw = out + (size_t)dN * DIM;

        #pragma unroll
        for (int c = 0; c < DIM; c += 32) {
            atomicAdd(&orw[c + lane], xr[c + lane] * p);
        }
    }
}

// ---------------------------------------------------------------------------
// Host launcher
// ---------------------------------------------------------------------------
extern "C" void kernel_launch(void* const* d_in, const int* in_sizes, int n_in,
                              void* d_out, int out_size, void* d_ws, size_t ws_size,
                              hipStream_t stream) {
    const float* x  = (const float*)d_in[0];   // [N, 256]
    const float* Wu = (const float*)d_in[1];   // [8, 256]
    const float* bu = (const float*)d_in[2];   // [8]
    const float* Wv = (const float*)d_in[3];   // [8, 256]
    const int* src  = (const int*)d_in[4];     // [E]
    const int* dst  = (const int*)d_in[5];     // [E]
    float* out = (float*)d_out;                // [N, 256]

    // Workspace layout (floats): s[N,16] | m[N,8] | denom[N,8] | e[E,8] = 32 MB
    float* ws    = (float*)d_ws;
    float* s     = ws;
    float* m     = s + (size_t)N_NODES * NCOL;
    float* denom = m + (size_t)N_NODES * NHEAD;
    float* e_ws  = denom + (size_t)N_NODES * NHEAD;

    gat_init_kernel<<<1024, 256, 0, stream>>>(out, m, denom);

    gat_score_gemm_kernel<<<(NTILES + 7) / 8, 256, 0, stream>>>(x, Wu, bu, Wv, s);

    gat_edge_logits_kernel<<<8192, 256, 0, stream>>>(s, src, dst, e_ws, m);

    gat_edge_exp_kernel<<<8192, 256, 0, stream>>>(dst, e_ws, m, denom);

    gat_scatter_kernel<<<8192, 256, 0, stream>>>(x, src, dst, e_ws, denom, out);
}